// RK4_block_lnRo_58110907515036
// MI455X (gfx1250) — compile-verified
//
#include <hip/hip_runtime.h>
#include <math.h>

// Problem constants from the reference
#define NKNOT 11        // M+1 hat-basis knots
#define NROW  50        // L*HID rows per knot
#define NTILE 4         // ceil(50/16) row tiles of 16
#define RHID  5
#define RL    10
#define STEPC 0.05f

typedef __attribute__((ext_vector_type(2))) float v2f;
typedef __attribute__((ext_vector_type(8))) float v8f;

// Native CDNA5 V_TANH_F32 when the builtin exists (TRANS op, co-executes with VALU/WMMA)
#if defined(__has_builtin)
#if __has_builtin(__builtin_amdgcn_tanhf)
#define FAST_TANHF(x) __builtin_amdgcn_tanhf(x)
#endif
#endif
#ifndef FAST_TANHF
#define FAST_TANHF(x) tanhf(x)
#endif

// hat (piecewise linear) basis weight at knot i for time t; T0=0, T1=1, M=10
__device__ __forceinline__ float hatphi(float t, int i) {
    float s  = t - 0.1f * (float)i;
    float r1 = fmaxf(s + 0.1f, 0.0f);
    float r2 = fmaxf(s,        0.0f);
    float r3 = fmaxf(s - 0.1f, 0.0f);
    return 10.0f * (r1 - 2.0f * r2 + r3);
}

__global__ __launch_bounds__(256) void rk4_divv_wmma(
    const float* __restrict__ lnRo,
    const float* __restrict__ x0,
    const float* __restrict__ x1,
    const float* __restrict__ x2,
    const float* __restrict__ t,
    const float* __restrict__ WW1,   // (11,10,5,2) -> flat row*2+d, row = i*50+l*5+h
    const float* __restrict__ WW2,   // (11,10,2,5)
    const float* __restrict__ bb1,   // (11,10,5)   -> flat row
    float* __restrict__ out,
    int R)
{
    __shared__ float s_ww1[NKNOT * NROW * 2]; // A-matrix source (row-major, 2 floats/row)
    __shared__ float s_bb1[NKNOT * NROW];     // bias per row
    __shared__ float s_wb [NKNOT * NROW];     // w_base[row] = sum_d WW2[i,l,d,h]*WW1[i,l,h,d]

    // ---- one-time per-block weight prep (550 rows, 8.8 KB LDS) ----
    // idx = il*5 + h with il = i*L + l; avoids div/rem magic-multiply chains.
    for (int il = (int)threadIdx.x; il < NKNOT * RL; il += (int)blockDim.x) {
        const int rowb = il * RHID;
        #pragma unroll
        for (int h = 0; h < RHID; ++h) {
            const int idx = rowb + h;
            float w10 = WW1[idx * 2 + 0];
            float w11 = WW1[idx * 2 + 1];
            s_ww1[idx * 2 + 0] = w10;
            s_ww1[idx * 2 + 1] = w11;
            s_bb1[idx] = bb1[idx];
            float w20 = WW2[(il * 2 + 0) * RHID + h];
            float w21 = WW2[(il * 2 + 1) * RHID + h];
            s_wb[idx] = w20 * w10 + w21 * w11;
        }
    }
    __syncthreads();

    const int  lane = threadIdx.x & 31;
    const int  wid  = threadIdx.x >> 5;
    const long base = ((long)blockIdx.x * (blockDim.x >> 5) + wid) * 16;
    const bool lo   = lane < 16;
    const int  col  = lane & 15;
    const int  hi8  = lo ? 0 : 8;            // C/D rows: M = v + 8*(lane>=16)
    const bool valid = lo && (base + col) < R;

    const float tt = t[0];
    const float te[3] = { tt, tt + STEPC * 0.5f, tt + STEPC };

    // B matrices (4x16 f32, K padded 2->4 with zeros): per eval, 16 samples/wave.
    // Layout mirrors documented 16x4 A layout: lanes 0-15 = K0(v0)/K1(v1), lanes 16-31 = K2/K3 (zero pad).
    v2f B[3];
    {
        const float* xs[3] = { x0, x1, x2 };
        #pragma unroll
        for (int e = 0; e < 3; ++e) {
            float bx = 0.f, by = 0.f;
            if (valid) {
                bx = xs[e][(base + col) * 2 + 0];
                by = xs[e][(base + col) * 2 + 1];
            }
            B[e].x = bx; B[e].y = by;
        }
    }

    float acc[3] = { 0.f, 0.f, 0.f };

    for (int i = 0; i < NKNOT; ++i) {
        // hat basis: at most 2 knots active per time; wave-uniform skip (t is scalar)
        const float p0 = hatphi(te[0], i);
        const float p1 = hatphi(te[1], i);
        const float p2 = hatphi(te[2], i);
        if (p0 == 0.f && p1 == 0.f && p2 == 0.f) continue;
        const float p[3] = { p0, p1, p2 };
        const int ib = i * NROW;

        #pragma unroll
        for (int tau = 0; tau < NTILE; ++tau) {
            const int rb = tau * 16;
            // A tile 16x4: lanes 0-15 carry WW1 row (K0,K1); K2,K3 halves are zero pad
            float a0 = 0.f, a1 = 0.f;
            if (lo && (rb + col) < NROW) {
                a0 = s_ww1[(ib + rb + col) * 2 + 0];
                a1 = s_ww1[(ib + rb + col) * 2 + 1];
            }
            v2f A; A.x = a0; A.y = a1;

            // bias (C input) and reduction weights in C/D register layout
            float cb[8], wv[8];
            #pragma unroll
            for (int v = 0; v < 8; ++v) {
                int  r  = rb + v + hi8;
                bool ok = r < NROW;
                cb[v] = ok ? s_bb1[ib + r] : 0.f;
                wv[v] = ok ? s_wb [ib + r] : 0.f;
            }
            v8f Cin = { cb[0], cb[1], cb[2], cb[3], cb[4], cb[5], cb[6], cb[7] };

            #pragma unroll
            for (int e = 0; e < 3; ++e) {
                if (p[e] == 0.f) continue;  // wave-uniform
                // z tile = WW1_tile (16x4) x X (4x16) + bias : one f32 WMMA
                v8f Z = __builtin_amdgcn_wmma_f32_16x16x4_f32(
                    false, A, false, B[e], (short)0, Cin, false, false);
                // tanh' weighted partial reduction over this tile's 8 rows
                float ts = 0.f;
                #pragma unroll
                for (int v = 0; v < 8; ++v) {
                    float th = FAST_TANHF(Z[v]);
                    ts = fmaf(wv[v], 1.0f - th * th, ts);
                }
                acc[e] = fmaf(p[e], ts, acc[e]);
            }
        }
    }

    // y = lnRo + step/6 * (-divV1 - 4*divV2 - divV3)
    float res = -(STEPC / 6.0f) * (acc[0] + 4.0f * acc[1] + acc[2]);
    // column n total = lane n partial + lane n+16 partial
    res += __shfl_xor(res, 16, 32);
    if (valid) {
        out[base + col] = lnRo[base + col] + res;
    }
}

extern "C" void kernel_launch(void* const* d_in, const int* in_sizes, int n_in,
                              void* d_out, int out_size, void* d_ws, size_t ws_size,
                              hipStream_t stream) {
    const float* lnRo = (const float*)d_in[0];
    const float* x0   = (const float*)d_in[1];
    const float* x1   = (const float*)d_in[2];
    const float* x2   = (const float*)d_in[3];
    const float* t    = (const float*)d_in[4];
    const float* WW1  = (const float*)d_in[5];
    const float* WW2  = (const float*)d_in[6];
    const float* bb1  = (const float*)d_in[7];
    float* out = (float*)d_out;

    const int R = in_sizes[0];                   // 131072
    const int samplesPerBlock = 16 * (256 / 32); // 8 waves x 16 samples = 128
    const int blocks = (R + samplesPerBlock - 1) / samplesPerBlock;
    rk4_divv_wmma<<<blocks, 256, 0, stream>>>(lnRo, x0, x1, x2, t, WW1, WW2, bb1, out, R);
}